// QuantumFeatureLayer_5446018531705
// MI455X (gfx1250) — compile-verified
//
#include <hip/hip_runtime.h>

// fp32 WMMA operand types for V_WMMA_F32_16X16X4_F32:
// A (16x4 f32)  = 64 f32 / 32 lanes = 2 VGPRs -> v2f
// B (4x16 f32)  = 64 f32 / 32 lanes = 2 VGPRs -> v2f
// C/D (16x16)   = 256 f32 / 32 lanes = 8 VGPRs -> v8f
typedef __attribute__((ext_vector_type(2))) float v2f;
typedef __attribute__((ext_vector_type(8))) float v8f;

#define QPI 3.14159f   // matches the reference's literal
#define HALF_QPI (0.5f * QPI)

// ---------------------------------------------------------------------------
// Setup: build the fixed 8x8 orthogonal matrix M implementing
// [RY(w[l,i]) layers + CNOT chains] by pushing the 8 basis vectors through.
// M is row-major: M[s*8 + t] = <s| U |t>, written into d_ws (256 bytes).
// ---------------------------------------------------------------------------
__global__ void qfl_build_M(const float* __restrict__ w, float* __restrict__ M) {
    if (threadIdx.x != 0 || blockIdx.x != 0) return;
    for (int t = 0; t < 8; ++t) {
        float a[8];
        #pragma unroll
        for (int s = 0; s < 8; ++s) a[s] = (s == t) ? 1.0f : 0.0f;
        for (int l = 0; l < 2; ++l) {
            // RY(w[l,q]) on wire q  (wire q is bit (2-q) of the flat index)
            for (int q = 0; q < 3; ++q) {
                float h = w[l * 3 + q] * 0.5f;
                float c = cosf(h), s = sinf(h);
                int stride = 1 << (2 - q);
                for (int i = 0; i < 8; ++i) {
                    if (i & stride) continue;
                    int j = i | stride;
                    float a0 = a[i], a1 = a[j];
                    a[i] = c * a0 - s * a1;
                    a[j] = s * a0 + c * a1;
                }
            }
            // CNOT(q, q+1): swap target bit where control bit == 1
            for (int q = 0; q < 2; ++q) {
                int cb = 1 << (2 - q);
                int tb = 1 << (1 - q);
                for (int i = 0; i < 8; ++i) {
                    if ((i & cb) && !(i & tb)) {
                        int j = i | tb;
                        float tmp = a[i]; a[i] = a[j]; a[j] = tmp;
                    }
                }
            }
        }
        for (int s = 0; s < 8; ++s) M[s * 8 + t] = a[s];
    }
}

// ---------------------------------------------------------------------------
// Main kernel: 256 threads = 8 waves; each wave handles 32 samples (1/lane).
// Per wave: W(8x32) = M(8x8) * V(8x32) via 4x v_wmma_f32_16x16x4_f32
// (2 sample groups x 2 K-chunks), then per-lane squares + signed sums.
// ---------------------------------------------------------------------------
__global__ __launch_bounds__(256) void qfl_main(const float4* __restrict__ x,
                                                const float* __restrict__ M,
                                                float* __restrict__ out) {
    __shared__ float vbuf[8][32 * 9];                 // stride-9 pad vs banks
    __shared__ __align__(16) float obuf[8][96];       // 32 samples x 3 outputs

    const int lane = threadIdx.x & 31;
    const int wave = threadIdx.x >> 5;
    const int sample = blockIdx.x * 256 + threadIdx.x;

    // ---- encoding: product state amplitudes v_k for this lane's sample ----
    float4 xv = x[sample];
    float s0, c0, s1, c1, s2, c2;
    sincosf(xv.x * HALF_QPI, &s0, &c0);
    sincosf(xv.y * HALF_QPI, &s1, &c1);
    sincosf(xv.z * HALF_QPI, &s2, &c2);

    float cc = c0 * c1, cs = c0 * s1, sc = s0 * c1, ss = s0 * s1;
    float* vs = vbuf[wave];
    const int vb = lane * 9;
    vs[vb + 0] = cc * c2;  vs[vb + 1] = cc * s2;
    vs[vb + 2] = cs * c2;  vs[vb + 3] = cs * s2;
    vs[vb + 4] = sc * c2;  vs[vb + 5] = sc * s2;
    vs[vb + 6] = ss * c2;  vs[vb + 7] = ss * s2;

    __syncthreads();

    // ---- gather WMMA operands ----
    const int r  = lane & 15;        // row (A) / column-sample (B)
    const int hi = lane >> 4;        // K half-select: K = base + hi*2
    const int rr = r & 7;            // clamp M row (rows 8..15 of A are zero)
    const float rowMask = (r < 8) ? 1.0f : 0.0f;

    // A operand: A[row, K] ; VGPR0 holds K={0|2}, VGPR1 K={1|3} per lane half
    v2f a0, a1;
    {
        const float* Mr = M + rr * 8 + hi * 2;
        a0.x = Mr[0] * rowMask;  a0.y = Mr[1] * rowMask;   // K-chunk 0..3
        a1.x = Mr[4] * rowMask;  a1.y = Mr[5] * rowMask;   // K-chunk 4..7
    }

    // B operand: B[K, n] = v_K(sample group_base + n), same K/K+2 lane split
    v2f b00, b01, b10, b11;
    {
        const float* p0 = vs + r * 9 + hi * 2;          // samples 0..15
        b00.x = p0[0];  b00.y = p0[1];
        b01.x = p0[4];  b01.y = p0[5];
        const float* p1 = vs + (16 + r) * 9 + hi * 2;   // samples 16..31
        b10.x = p1[0];  b10.y = p1[1];
        b11.x = p1[4];  b11.y = p1[5];
    }

    // ---- 4x fp32 WMMA: D[s, n] = sum_k M[s,k] * v_k(n) ----
    v8f d0 = {};
    d0 = __builtin_amdgcn_wmma_f32_16x16x4_f32(false, a0, false, b00,
                                               (short)0, d0, false, false);
    d0 = __builtin_amdgcn_wmma_f32_16x16x4_f32(false, a1, false, b01,
                                               (short)0, d0, false, false);
    v8f d1 = {};
    d1 = __builtin_amdgcn_wmma_f32_16x16x4_f32(false, a0, false, b10,
                                               (short)0, d1, false, false);
    d1 = __builtin_amdgcn_wmma_f32_16x16x4_f32(false, a1, false, b11,
                                               (short)0, d1, false, false);

    // ---- D layout: lanes 0..15, VGPR g = amplitude state g of sample lane --
    float* os = obuf[wave];
    if (lane < 16) {
        float p[8];
        #pragma unroll
        for (int s = 0; s < 8; ++s) { float t = d0[s]; p[s] = t * t; }
        os[lane * 3 + 0] = (p[0]+p[1]+p[2]+p[3]) - (p[4]+p[5]+p[6]+p[7]);
        os[lane * 3 + 1] = (p[0]+p[1]+p[4]+p[5]) - (p[2]+p[3]+p[6]+p[7]);
        os[lane * 3 + 2] = (p[0]+p[2]+p[4]+p[6]) - (p[1]+p[3]+p[5]+p[7]);
        #pragma unroll
        for (int s = 0; s < 8; ++s) { float t = d1[s]; p[s] = t * t; }
        os[(16 + lane) * 3 + 0] = (p[0]+p[1]+p[2]+p[3]) - (p[4]+p[5]+p[6]+p[7]);
        os[(16 + lane) * 3 + 1] = (p[0]+p[1]+p[4]+p[5]) - (p[2]+p[3]+p[6]+p[7]);
        os[(16 + lane) * 3 + 2] = (p[0]+p[2]+p[4]+p[6]) - (p[1]+p[3]+p[5]+p[7]);
    }
    __syncthreads();

    // ---- coalesced store: 96 floats per wave = 24 float4 stores ----
    const size_t waveBase = (size_t)blockIdx.x * 256 + (size_t)wave * 32;
    if (lane < 24) {
        float4* dst = (float4*)(out + waveBase * 3);
        dst[lane] = ((const float4*)os)[lane];
    }
}

extern "C" void kernel_launch(void* const* d_in, const int* in_sizes, int n_in,
                              void* d_out, int out_size, void* d_ws, size_t ws_size,
                              hipStream_t stream) {
    const float* x = (const float*)d_in[0];    // (B, 4) float32
    const float* w = (const float*)d_in[1];    // (2, 3) float32
    float* M = (float*)d_ws;                   // 64 floats scratch
    float* out = (float*)d_out;                // (B, 3) float32

    const int B = in_sizes[0] / 4;             // 2,097,152

    qfl_build_M<<<1, 64, 0, stream>>>(w, M);
    // B is a multiple of 256 (2^21): full blocks, EXEC all-ones at WMMAs.
    qfl_main<<<B / 256, 256, 0, stream>>>((const float4*)x, M, out);
}